// Wav2vec2Loss_19490561590149
// MI455X (gfx1250) — compile-verified
//
#include <hip/hip_runtime.h>
#include <math.h>

typedef __attribute__((ext_vector_type(2))) float v2f;
typedef __attribute__((ext_vector_type(8))) float v8f;

#define BB     16
#define TT     400
#define DD     512
#define MM     100
#define PAD_M  112            // 7*16, zero-padded rows/cols per batch
#define NTILE  7              // PAD_M / 16
#define NROWS  (BB * MM)      // 1600
#define TEMP_INV 10.0f        // 1/0.1
#define ALPHA  0.1f
#define EPSN   1e-8f
#define GV     640            // G*V = 2*320

// ---------------------------------------------------------------------------
// K1: gather masked rows and L2-normalize; pad rows [100..111] with zeros so
// the WMMA GEMM needs no masking (EXEC must be all-1s around WMMA) and the
// padded region of S is exactly zero.
// grid.x = BB*PAD_M, block = 128 (each thread owns one float4 of the row)
// ---------------------------------------------------------------------------
__global__ void gather_normalize_kernel(const float* __restrict__ src,
                                        const int*   __restrict__ mask,
                                        float*       __restrict__ dst) {
    int row = blockIdx.x;            // 0 .. BB*PAD_M-1
    int b   = row / PAD_M;
    int r   = row % PAD_M;
    int t   = threadIdx.x;           // 0..127
    float* out = dst + (size_t)row * DD;

    if (r >= MM) {                   // zero padding row (uniform per block)
        ((float4*)out)[t] = make_float4(0.f, 0.f, 0.f, 0.f);
        return;
    }

    int tidx = mask[b * MM + r];
    const float* in = src + ((size_t)b * TT + (size_t)tidx) * DD;
    float4 v = ((const float4*)in)[t];

    float ss = v.x*v.x + v.y*v.y + v.z*v.z + v.w*v.w;
    __shared__ float red[128];
    red[t] = ss;
    __syncthreads();
    for (int off = 64; off > 0; off >>= 1) {
        if (t < off) red[t] += red[t + off];
        __syncthreads();
    }
    float inv = 1.0f / fmaxf(sqrtf(red[0]), EPSN);
    v.x *= inv; v.y *= inv; v.z *= inv; v.w *= inv;
    ((float4*)out)[t] = v;
}

// ---------------------------------------------------------------------------
// K2: per-batch similarity GEMM  S_b = Tn_b (112x512) * Lhat_b^T (512x112)
// via V_WMMA_F32_16X16X4_F32. One wave32 owns a 16x112 strip: the A fragment
// is loaded once per k-step and reused across 7 WMMAs (7 independent
// accumulator chains -> ILP to hide L2 latency).
// A layout (ISA 7.12.2, 32-bit A 16x4): lane = 16*hi + lo holds
//   a = { A[lo][k0+2*hi], A[lo][k0+2*hi+1] }
// B = Lhat^T, so b = { Lhat[tile_n*16+lo][k0+2*hi], ...[k+1] } (contiguous).
// C/D layout: VGPR r -> row r + 8*hi, col lo. Stores are branchless into the
// fully padded S (pad entries are 0*0 products).
// grid = (7 tm, 16 b), block = 32
// ---------------------------------------------------------------------------
__global__ void gemm_sim_kernel(const float* __restrict__ Tn,
                                const float* __restrict__ Lh,
                                float*       __restrict__ S) {
    int lane = threadIdx.x;          // 0..31
    int lo   = lane & 15;
    int hi   = lane >> 4;
    int tm   = blockIdx.x;           // 0..6
    int b    = blockIdx.y;           // 0..15

    const float* Arow = Tn + (size_t)(b * PAD_M + tm * 16 + lo) * DD + 2 * hi;
    const float* Bbase = Lh + (size_t)(b * PAD_M + lo) * DD + 2 * hi;

    v8f c[NTILE];
    #pragma unroll
    for (int t = 0; t < NTILE; ++t) c[t] = (v8f){};

    #pragma unroll 4
    for (int k0 = 0; k0 < DD; k0 += 4) {
        v2f a = *(const v2f*)(Arow + k0);            // 8B-aligned float2
        #pragma unroll
        for (int t = 0; t < NTILE; ++t) {
            v2f bb = *(const v2f*)(Bbase + (size_t)t * 16 * DD + k0);
            c[t] = __builtin_amdgcn_wmma_f32_16x16x4_f32(
                       /*neg_a=*/false, a, /*neg_b=*/false, bb,
                       /*c_mod=*/(short)0, c[t],
                       /*reuse_a=*/false, /*reuse_b=*/false);
        }
    }

    // Branchless store of the 16x112 strip into padded S.
    float* Srow = S + (size_t)(b * PAD_M + tm * 16) * PAD_M;
    #pragma unroll
    for (int t = 0; t < NTILE; ++t) {
        int gn = t * 16 + lo;
        #pragma unroll
        for (int r = 0; r < 8; ++r) {
            int rr = r + 8 * hi;                      // row within strip
            Srow[(size_t)rr * PAD_M + gn] = c[t][r];
        }
    }
}

// ---------------------------------------------------------------------------
// K3: per-sample row statistics using the actual neg_indices input.
// Entry 0 = self (label n), entries 1..99 = neg_indices[n][*] (in-batch).
// Produces sim[n], expsum[n] = sum exp(s/T), linsum[n] = sum s.
// grid = NROWS, block = 128
// ---------------------------------------------------------------------------
__global__ void rowstats_kernel(const float* __restrict__ S,
                                const int*   __restrict__ negIdx,
                                float* __restrict__ simA,
                                float* __restrict__ expA,
                                float* __restrict__ linA) {
    int n = blockIdx.x;              // 0..1599
    int b = n / MM;
    int j = n - b * MM;
    int t = threadIdx.x;             // 0..127
    const float* Srow = S + (size_t)(b * PAD_M + j) * PAD_M;

    float s = 0.f, e = 0.f;
    if (t < MM) {
        int col = (t == 0) ? j : (negIdx[(size_t)n * (MM - 1) + (t - 1)] - b * MM);
        float v = Srow[col];
        s = v;
        e = expf(v * TEMP_INV);
    }
    __shared__ float rs[128], re[128];
    rs[t] = s; re[t] = e;
    __syncthreads();
    for (int off = 64; off > 0; off >>= 1) {
        if (t < off) { rs[t] += rs[t + off]; re[t] += re[t + off]; }
        __syncthreads();
    }
    if (t == 0) {
        simA[n] = Srow[j];
        linA[n] = rs[0];
        expA[n] = re[0];
    }
}

// ---------------------------------------------------------------------------
// K4: deterministic single-block finalize -> 5 scalar outputs
// (loss, contrastive, diversity, pos_sim, neg_sim_mean)
// ---------------------------------------------------------------------------
__global__ void finalize_kernel(const float* __restrict__ simA,
                                const float* __restrict__ expA,
                                const float* __restrict__ linA,
                                const float* __restrict__ perp,
                                float* __restrict__ out) {
    int t = threadIdx.x;             // 0..255
    float cl = 0.f, ps = 0.f, ns = 0.f, dv = 0.f;
    for (int n = t; n < NROWS; n += 256) {
        float sim = simA[n];
        cl += logf(expA[n]) - sim * TEMP_INV;
        ps += sim;
        ns += (linA[n] - sim) * (1.0f / (MM - 1));
    }
    for (int i = t; i < GV; i += 256) {
        float p = perp[i];
        dv += p * logf(p);
    }
    __shared__ float r0[256], r1[256], r2[256], r3[256];
    r0[t] = cl; r1[t] = ps; r2[t] = ns; r3[t] = dv;
    __syncthreads();
    for (int off = 128; off > 0; off >>= 1) {
        if (t < off) {
            r0[t] += r0[t + off]; r1[t] += r1[t + off];
            r2[t] += r2[t + off]; r3[t] += r3[t + off];
        }
        __syncthreads();
    }
    if (t == 0) {
        float contrastive = r0[0] / (float)NROWS;
        float pos         = r1[0] / (float)NROWS;
        float negm        = r2[0] / (float)NROWS;
        float diversity   = r3[0] / (float)GV;
        out[0] = contrastive + ALPHA * diversity;  // loss
        out[1] = contrastive;
        out[2] = diversity;
        out[3] = pos;
        out[4] = negm;
    }
}

// ---------------------------------------------------------------------------
extern "C" void kernel_launch(void* const* d_in, const int* in_sizes, int n_in,
                              void* d_out, int out_size, void* d_ws, size_t ws_size,
                              hipStream_t stream) {
    const float* enc    = (const float*)d_in[0];  // (B,T,D) f32
    const float* quant  = (const float*)d_in[1];  // (B,T,D) f32
    const float* perp   = (const float*)d_in[2];  // (G,V)   f32
    const int*   mask   = (const int*)  d_in[3];  // (B,M)   int
    const int*   negIdx = (const int*)  d_in[4];  // (B*M, M-1) int
    float* out = (float*)d_out;

    // Workspace carve-out (~8.2 MB of floats), fully rewritten every call.
    float* ws = (float*)d_ws;
    size_t o = 0;
    float* Tn   = ws + o; o += (size_t)BB * PAD_M * DD;     // 917504
    float* Lh   = ws + o; o += (size_t)BB * PAD_M * DD;     // 917504
    float* S    = ws + o; o += (size_t)BB * PAD_M * PAD_M;  // 200704
    float* simA = ws + o; o += NROWS;
    float* expA = ws + o; o += NROWS;
    float* linA = ws + o; o += NROWS;

    gather_normalize_kernel<<<dim3(BB * PAD_M), 128, 0, stream>>>(enc,   mask, Tn);
    gather_normalize_kernel<<<dim3(BB * PAD_M), 128, 0, stream>>>(quant, mask, Lh);
    gemm_sim_kernel<<<dim3(NTILE, BB), 32, 0, stream>>>(Tn, Lh, S);
    rowstats_kernel<<<dim3(NROWS), 128, 0, stream>>>(S, negIdx, simA, expA, linA);
    finalize_kernel<<<dim3(1), 256, 0, stream>>>(simA, expA, linA, perp, out);
}